// MixActivConv2d_87617332839037
// MI455X (gfx1250) — compile-verified
//
#include <hip/hip_runtime.h>

typedef float v2f __attribute__((ext_vector_type(2)));
typedef float v4f __attribute__((ext_vector_type(4)));
typedef float v8f __attribute__((ext_vector_type(8)));

#define C_IN   256
#define C_OUT  256
#define HW     4096
#define BATCH  32
#define NSEL   8

#define MT 128
#define NT 128
#define KT 32
#define AP 36     // A LDS row stride (floats)
#define BPR 288   // B LDS k-pair row stride (floats): 128 cols * 2 + 32 pad

// ws layout (in 4-byte units)
// [0..3]   : encoded atomic min/max slots (sel_min, sel_max, w_min, w_max)
// [8..14]  : activation quant params: mn, c0,c1,c2, r0,r1,r2
// [16..79] : per-channel selected mask, packed uchar[256]
// [1024..] : quantized weight, 256x256 fp32 row-major (oc, ic)

__device__ __forceinline__ unsigned f32_key(float x) {
  unsigned b = __float_as_uint(x);
  return (b & 0x80000000u) ? ~b : (b | 0x80000000u);
}
__device__ __forceinline__ float key_f32(unsigned k) {
  return __uint_as_float((k & 0x80000000u) ? (k ^ 0x80000000u) : ~k);
}

// CDNA5 async copy: global -> LDS, 16B per lane, tracked by ASYNCcnt.
__device__ __forceinline__ void async_b128_to_lds(const void* gaddr, void* lds) {
  unsigned loff = (unsigned)(unsigned long long)lds;           // low 32 bits = LDS offset
  unsigned long long ga = (unsigned long long)gaddr;
  asm volatile("global_load_async_to_lds_b128 %0, %1, off"
               :: "v"(loff), "v"(ga) : "memory");
}
__device__ __forceinline__ void wait_asynccnt0() {
  asm volatile("s_wait_asynccnt 0" ::: "memory");
}

__global__ void k_init(const int* __restrict__ selch, unsigned* __restrict__ wsu,
                       unsigned char* __restrict__ selmask) {
  int t = threadIdx.x;
  if (t == 0) { wsu[0] = 0xFFFFFFFFu; wsu[2] = 0xFFFFFFFFu; }  // min slots
  if (t == 1) { wsu[1] = 0u;          wsu[3] = 0u;          }  // max slots
  if (t < C_IN) selmask[t] = 0;
  __syncthreads();
  if (t < NSEL) selmask[selch[t]] = 1;
}

__global__ void k_minmax_sel(const float* __restrict__ x, const int* __restrict__ selch,
                             unsigned* __restrict__ wsu) {
  __shared__ float smn[256], smx[256];
  __shared__ int sc[NSEL];
  int t = threadIdx.x;
  if (t < NSEL) sc[t] = selch[t];
  __syncthreads();
  float mn =  __int_as_float(0x7F800000);   // +inf
  float mx = -__int_as_float(0x7F800000);
  const long total = (long)BATCH * NSEL * HW;
  for (long i = (long)blockIdx.x * blockDim.x + t; i < total;
       i += (long)gridDim.x * blockDim.x) {
    int hw   = (int)(i % HW);
    int rest = (int)(i / HW);
    int s    = rest % NSEL;
    int b    = rest / NSEL;
    float v = x[((long)(b * C_IN + sc[s])) * HW + hw];
    mn = fminf(mn, v); mx = fmaxf(mx, v);
  }
  smn[t] = mn; smx[t] = mx; __syncthreads();
  for (int off = 128; off; off >>= 1) {
    if (t < off) { smn[t] = fminf(smn[t], smn[t+off]); smx[t] = fmaxf(smx[t], smx[t+off]); }
    __syncthreads();
  }
  if (t == 0) {
    atomicMin(&wsu[0], f32_key(smn[0]));
    atomicMax(&wsu[1], f32_key(smx[0]));
  }
}

__global__ void k_minmax_w(const float* __restrict__ w, unsigned* __restrict__ wsu) {
  __shared__ float smn[256], smx[256];
  int t = threadIdx.x;
  float mn =  __int_as_float(0x7F800000);
  float mx = -__int_as_float(0x7F800000);
  const int total = C_OUT * C_IN;
  for (int i = blockIdx.x * blockDim.x + t; i < total; i += gridDim.x * blockDim.x) {
    float v = w[i];
    mn = fminf(mn, v); mx = fmaxf(mx, v);
  }
  smn[t] = mn; smx[t] = mx; __syncthreads();
  for (int off = 128; off; off >>= 1) {
    if (t < off) { smn[t] = fminf(smn[t], smn[t+off]); smx[t] = fmaxf(smx[t], smx[t+off]); }
    __syncthreads();
  }
  if (t == 0) {
    atomicMin(&wsu[2], f32_key(smn[0]));
    atomicMax(&wsu[3], f32_key(smx[0]));
  }
}

__global__ void k_quant_w(const float* __restrict__ w, const float* __restrict__ alpha_a,
                          const float* __restrict__ alpha_w, const unsigned* __restrict__ wsu,
                          float* __restrict__ params, float* __restrict__ wq) {
  int idx = blockIdx.x * blockDim.x + threadIdx.x;
  float a0 = alpha_w[0], a1 = alpha_w[1], a2 = alpha_w[2];
  float m  = fmaxf(a0, fmaxf(a1, a2));
  float e0 = __expf(a0 - m), e1 = __expf(a1 - m), e2 = __expf(a2 - m);
  float inv = 1.0f / (e0 + e1 + e2);
  float sw0 = e0 * inv, sw1 = e1 * inv, sw2 = e2 * inv;
  float wmn = key_f32(wsu[2]);
  float wmx = key_f32(wsu[3]);
  float s0 = (wmx - wmn) / 3.0f, s1 = (wmx - wmn) / 15.0f, s2 = (wmx - wmn) / 255.0f;
  float v  = w[idx];
  float xm = v - wmn;
  // softmax sums to 1 => mixture = mn + sum_i sw_i*scale_i*round((v-mn)/scale_i)
  wq[idx] = wmn + sw0 * s0 * rintf(xm / s0)
                + sw1 * s1 * rintf(xm / s1)
                + sw2 * s2 * rintf(xm / s2);
  if (idx == 0) {
    float b0 = alpha_a[0], b1 = alpha_a[1], b2 = alpha_a[2];
    float mm = fmaxf(b0, fmaxf(b1, b2));
    float f0 = __expf(b0 - mm), f1 = __expf(b1 - mm), f2 = __expf(b2 - mm);
    float fi = 1.0f / (f0 + f1 + f2);
    float t0 = f0 * fi, t1 = f1 * fi, t2 = f2 * fi;
    float amn = key_f32(wsu[0]);
    float amx = key_f32(wsu[1]);
    float q0 = (amx - amn) / 3.0f, q1 = (amx - amn) / 15.0f, q2 = (amx - amn) / 255.0f;
    params[0] = amn;
    params[1] = t0 * q0; params[2] = t1 * q1; params[3] = t2 * q2;
    params[4] = 1.0f / q0; params[5] = 1.0f / q1; params[6] = 1.0f / q2;
  }
}

__global__ __launch_bounds__(256)
void k_gemm(const float* __restrict__ x, const float* __restrict__ wq,
            const float* __restrict__ params, const unsigned char* __restrict__ selmask,
            float* __restrict__ out) {
  __shared__ float Als[2][MT * AP];         // 2 x 18432 B, double-buffered (async W copies)
  __shared__ float Bls[(KT / 2) * BPR];     // 18432 B, k-pair interleaved

  const int t    = threadIdx.x;
  const int lane = t & 31;
  const int wv   = t >> 5;
  const int wm   = wv & 3;          // 4 waves along M (32 rows each)
  const int wn   = wv >> 2;         // 2 waves along N (64 cols each)
  const int n0   = blockIdx.x * NT;
  const int m0   = blockIdx.y * MT;
  const int b    = blockIdx.z;

  const float p0 = params[0], p1 = params[1], p2 = params[2], p3 = params[3];
  const float p4 = params[4], p5 = params[5], p6 = params[6];

  v8f acc[2][4];
  const v8f zacc = {0.f,0.f,0.f,0.f,0.f,0.f,0.f,0.f};
#pragma unroll
  for (int f = 0; f < 2; ++f)
#pragma unroll
    for (int g = 0; g < 4; ++g) acc[f][g] = zacc;

  // W staging: rows wr + i*32, one b128 async per row group
  const int wr  = t >> 3;
  const int wc4 = (t & 7) * 4;
  // X staging: per-thread 4-row column segments (K-pairs adjacent in regs)
  const int col = t & 127;           // column within N tile
  const int kk0 = (t >> 7) * 4;      // 0 or 4; rows kk0+8i .. +3

  // --- per-thread base pointers; all hot-loop accesses use constant offsets ---
  const float* xptr = x + ((long)b * C_IN + kk0) * HW + n0 + col;  // += KT*HW per chunk
  const float* wptr = wq + (m0 + wr) * C_IN + wc4;                 // += KT per chunk
  const unsigned char* mptr = selmask + kk0;                       // += KT per chunk

  float xv[4][4];
  unsigned msk4[4];

  // prologue: async W chunk 0 -> Als[0]; prefetch X chunk 0 -> regs
#pragma unroll
  for (int i = 0; i < 4; ++i)
    async_b128_to_lds(&wptr[i * 32 * C_IN], &Als[0][(wr + i * 32) * AP + wc4]);
#pragma unroll
  for (int i = 0; i < 4; ++i) {
    msk4[i] = *(const unsigned*)&mptr[8 * i];
#pragma unroll
    for (int j = 0; j < 4; ++j)
      xv[i][j] = xptr[(8 * i + j) * HW];
  }

  int buf = 0;
  for (int kc = 0; kc < C_IN / KT; ++kc) {
    __syncthreads();    // all waves done reading previous chunk's LDS
    // stage X regs -> LDS; fuse activation fake-quant on selected channels
#pragma unroll
    for (int i = 0; i < 4; ++i) {
      if (msk4[i]) {
#pragma unroll
        for (int j = 0; j < 4; ++j) {
          if ((msk4[i] >> (8 * j)) & 1u) {
            float xm = xv[i][j] - p0;
            xv[i][j] = p0 + p1 * rintf(xm * p4) + p2 * rintf(xm * p5) + p3 * rintf(xm * p6);
          }
        }
      }
      const int p = (kk0 + 8 * i) >> 1;   // k-pair row
      v2f lo; lo.x = xv[i][0]; lo.y = xv[i][1];
      v2f hi; hi.x = xv[i][2]; hi.y = xv[i][3];
      *(v2f*)&Bls[p * BPR + 2 * col]       = lo;
      *(v2f*)&Bls[(p + 1) * BPR + 2 * col] = hi;
    }
    wait_asynccnt0();   // this wave's async W copy into Als[buf] has landed
    __syncthreads();    // all waves' B stores + W async copies visible

    if (kc + 1 < C_IN / KT) {
      xptr += KT * HW;
      wptr += KT;
      mptr += KT;
      // async W for next chunk into the other A buffer (overlaps compute below)
#pragma unroll
      for (int i = 0; i < 4; ++i)
        async_b128_to_lds(&wptr[i * 32 * C_IN], &Als[buf ^ 1][(wr + i * 32) * AP + wc4]);
#pragma unroll
      for (int i = 0; i < 4; ++i) {
        msk4[i] = *(const unsigned*)&mptr[8 * i];
#pragma unroll
        for (int j = 0; j < 4; ++j)
          xv[i][j] = xptr[(8 * i + j) * HW];
      }
    }

    // compute: KT=32 as 8 K-steps of 4; all LDS reads = per-lane base + immediate
    const float* Ab = &Als[buf][0];
    const int ml   = lane & 15;
    const int hi16 = lane >> 4;                         // lanes 16-31: K+2,K+3
    const int Afb  = (wm * 32 + ml) * AP + (hi16 << 1); // A frag base (floats)
    const int Bfb  = hi16 * BPR + 2 * (wn * 64 + ml);   // B frag base (floats)
#pragma unroll
    for (int ks = 0; ks < KT / 4; ++ks) {
      v2f a0 = *(const v2f*)&Ab[Afb + ks * 4];
      v2f a1 = *(const v2f*)&Ab[Afb + 16 * AP + ks * 4];
#pragma unroll
      for (int g = 0; g < 4; ++g) {
        v2f bb = *(const v2f*)&Bls[Bfb + ks * 2 * BPR + g * 32];
        acc[0][g] = __builtin_amdgcn_wmma_f32_16x16x4_f32(
            false, a0, false, bb, (short)0, acc[0][g], false, false);
        acc[1][g] = __builtin_amdgcn_wmma_f32_16x16x4_f32(
            false, a1, false, bb, (short)0, acc[1][g], false, false);
      }
    }
    buf ^= 1;
  }

  // store C/D: VGPR i -> lanes 0-15: M=i, lanes 16-31: M=8+i
  // single base pointer + constant offsets
  const int ml    = lane & 15;
  const int mhalf = (lane >> 4) << 3;
  float* op = out + ((long)b * C_OUT + m0 + wm * 32 + mhalf) * HW + n0 + wn * 64 + ml;
#pragma unroll
  for (int f = 0; f < 2; ++f) {
#pragma unroll
    for (int g = 0; g < 4; ++g) {
#pragma unroll
      for (int i = 0; i < 8; ++i) {
        op[(f * 16 + i) * HW + g * 16] = acc[f][g][i];
      }
    }
  }
}

extern "C" void kernel_launch(void* const* d_in, const int* in_sizes, int n_in,
                              void* d_out, int out_size, void* d_ws, size_t ws_size,
                              hipStream_t stream) {
  const float* x       = (const float*)d_in[0];
  const float* alpha_a = (const float*)d_in[1];
  const float* alpha_w = (const float*)d_in[2];
  const float* w       = (const float*)d_in[3];
  const int*   selch   = (const int*)d_in[4];
  float* out = (float*)d_out;

  unsigned* wsu = (unsigned*)d_ws;
  float*    wsf = (float*)d_ws;
  float* params          = wsf + 8;
  unsigned char* selmask = (unsigned char*)d_ws + 64;
  float* wqbuf           = wsf + 1024;     // 256x256 fp32 quantized weight

  k_init<<<1, 256, 0, stream>>>(selch, wsu, selmask);
  k_minmax_sel<<<512, 256, 0, stream>>>(x, selch, wsu);
  k_minmax_w<<<64, 256, 0, stream>>>(w, wsu);
  k_quant_w<<<(C_OUT * C_IN) / 256, 256, 0, stream>>>(w, alpha_a, alpha_w, wsu, params, wqbuf);

  dim3 grid(HW / NT, C_OUT / MT, BATCH);
  k_gemm<<<grid, 256, 0, stream>>>(x, wqbuf, params, selmask, out);
}